// BilateralFilter_6468220748252
// MI455X (gfx1250) — compile-verified
//
#include <hip/hip_runtime.h>
#include <hip/hip_bf16.h>
#include <math.h>

// ---------------------------------------------------------------------------
// Joint bilateral filter, 3x3 window, zero padding.
//   depth [N,1,512,512], color [N,3,512,512], mask [N,1,512,512]  (N = B*V = 8)
//   out   [N,3,512,512]
// Single pass per window using:
//   u(t)   = wd(t)*wc(t)*ws(t)*mp(t)
//   S      = (sum wd) * (sum wc)
//   out_c  = (sum u*cp_c + 1e-7*S*sum cp_c) / (sum u + 9e-7*S)
// which equals the reference's normalized formulation exactly.
// ---------------------------------------------------------------------------

#define IMG_H 512
#define IMG_W 512
#define TW 32
#define THT 32
#define LW (TW + 2)          // 34
#define LH (THT + 2)         // 34
#define NSLOTS (LW * LH)     // 1156
#define NTHREADS 256

// INV2S2 = 0.78125 exactly (sigma = 0.8).  NEGK2 = -INV2S2 * log2(e).
#define NEGK2 (-1.1271055006945027f)
// Normalized 3x3 spatial gaussian (exp(-(dx^2+dy^2)*0.78125), sum-normalized).
#define WS_C 0.27249635f     // center
#define WS_E 0.12475766f     // edge
#define WS_K 0.05711816f     // corner

#if defined(__AMDGCN__) && defined(__has_builtin)
#  if __has_builtin(__builtin_amdgcn_global_load_async_to_lds_b32) && \
      __has_builtin(__builtin_amdgcn_s_wait_asynccnt)
#    define HAVE_ASYNC_LDS 1
#  endif
#endif
#ifndef HAVE_ASYNC_LDS
#  define HAVE_ASYNC_LDS 0
#endif

#if defined(__AMDGCN__)
#  define FAST_EX2(x) __builtin_amdgcn_exp2f(x)
#  define FAST_RCP(x) __builtin_amdgcn_rcpf(x)
#else
#  define FAST_EX2(x) exp2f(x)
#  define FAST_RCP(x) (1.0f / (x))
#endif

#if HAVE_ASYNC_LDS
typedef __attribute__((address_space(1))) int* g_i32p;
typedef __attribute__((address_space(3))) int* l_i32p;
#  define ASYNC_G2L_B32(ldsp, gp)                                             \
    __builtin_amdgcn_global_load_async_to_lds_b32(                            \
        (g_i32p)(gp), (l_i32p)(ldsp), 0, 0)
#endif

__global__ __launch_bounds__(NTHREADS) void bilateral3x3_kernel(
    const float* __restrict__ depth,   // [N,1,H,W]
    const float* __restrict__ color,   // [N,3,H,W]
    const float* __restrict__ mask,    // [N,1,H,W]
    float* __restrict__ out)           // [N,3,H,W]
{
    __shared__ float sD[NSLOTS];
    __shared__ float sM[NSLOTS];
    __shared__ float sC0[NSLOTS];
    __shared__ float sC1[NSLOTS];
    __shared__ float sC2[NSLOTS];

    const int n   = blockIdx.z;
    const int tx0 = blockIdx.x * TW;
    const int ty0 = blockIdx.y * THT;
    const int tid = threadIdx.x;

    const size_t plane = (size_t)IMG_H * IMG_W;
    const float* gD  = depth + (size_t)n * plane;
    const float* gM  = mask  + (size_t)n * plane;
    const float* gC0 = color + (size_t)(3 * n)     * plane;
    const float* gC1 = gC0 + plane;
    const float* gC2 = gC1 + plane;

    // ---- Stage haloed 34x34 tile (5 planes) into LDS ----------------------
    for (int idx = tid; idx < NSLOTS; idx += NTHREADS) {
        const int ly = idx / LW;
        const int lx = idx - ly * LW;
        const int gy = ty0 + ly - 1;
        const int gx = tx0 + lx - 1;
        // Clamp so async loads never fault; OOB slots are zeroed afterwards.
        const int cy = min(max(gy, 0), IMG_H - 1);
        const int cx = min(max(gx, 0), IMG_W - 1);
        const size_t off = (size_t)cy * IMG_W + cx;
#if HAVE_ASYNC_LDS
        ASYNC_G2L_B32(&sD[idx],  gD  + off);
        ASYNC_G2L_B32(&sM[idx],  gM  + off);
        ASYNC_G2L_B32(&sC0[idx], gC0 + off);
        ASYNC_G2L_B32(&sC1[idx], gC1 + off);
        ASYNC_G2L_B32(&sC2[idx], gC2 + off);
#else
        const bool ib = (gy == cy) && (gx == cx);
        sD[idx]  = ib ? gD[off]  : 0.0f;
        sM[idx]  = ib ? gM[off]  : 0.0f;
        sC0[idx] = ib ? gC0[off] : 0.0f;
        sC1[idx] = ib ? gC1[off] : 0.0f;
        sC2[idx] = ib ? gC2[off] : 0.0f;
#endif
    }

#if HAVE_ASYNC_LDS
    __builtin_amdgcn_s_wait_asynccnt(0);
    __syncthreads();
    // Zero-padding fixup: only image-border tiles actually store anything.
    for (int idx = tid; idx < NSLOTS; idx += NTHREADS) {
        const int ly = idx / LW;
        const int lx = idx - ly * LW;
        const int gy = ty0 + ly - 1;
        const int gx = tx0 + lx - 1;
        if ((unsigned)gy >= (unsigned)IMG_H || (unsigned)gx >= (unsigned)IMG_W) {
            sD[idx] = 0.0f; sM[idx] = 0.0f;
            sC0[idx] = 0.0f; sC1[idx] = 0.0f; sC2[idx] = 0.0f;
        }
    }
#endif
    __syncthreads();

    // ---- Compute: each thread does 4 pixels (rows py0, py0+8, +16, +24) ---
    const float wsTab[9] = { WS_K, WS_E, WS_K,
                             WS_E, WS_C, WS_E,
                             WS_K, WS_E, WS_K };

    const int px  = tid & 31;   // 0..31, lane-contiguous -> coalesced stores
    const int py0 = tid >> 5;   // 0..7

    float* oBase = out + (size_t)(3 * n) * plane;

#pragma unroll
    for (int k = 0; k < 4; ++k) {
        const int py = py0 + 8 * k;
        const int ci = (py + 1) * LW + (px + 1);

        const float d  = sD[ci];
        const float c0 = sC0[ci];
        const float c1 = sC1[ci];
        const float c2 = sC2[ci];

        float sum_u = 0.0f, swd = 0.0f, swc = 0.0f;
        float su0 = 0.0f, su1 = 0.0f, su2 = 0.0f;
        float sc0 = 0.0f, sc1 = 0.0f, sc2 = 0.0f;

#pragma unroll
        for (int t = 0; t < 9; ++t) {
            const int dy = t / 3 - 1;
            const int dx = t % 3 - 1;
            const int ti = ci + dy * LW + dx;

            const float dp = sD[ti];
            const float mp = sM[ti];
            const float p0 = sC0[ti];
            const float p1 = sC1[ti];
            const float p2 = sC2[ti];

            const float dd = dp - d;
            const float wd = FAST_EX2(dd * dd * NEGK2);

            const float e0 = p0 - c0;
            const float e1 = p1 - c1;
            const float e2 = p2 - c2;
            const float wc = FAST_EX2(e0 * e0 * NEGK2)
                           + FAST_EX2(e1 * e1 * NEGK2)
                           + FAST_EX2(e2 * e2 * NEGK2);

            const float u = wd * wc * wsTab[t] * mp;

            sum_u += u;  swd += wd;  swc += wc;
            su0 += u * p0;  su1 += u * p1;  su2 += u * p2;
            sc0 += p0;      sc1 += p1;      sc2 += p2;
        }

        const float S    = swd * swc;
        const float eps  = 1e-7f * S;
        const float rden = FAST_RCP(sum_u + 9.0f * eps);

        const size_t o = (size_t)(ty0 + py) * IMG_W + (tx0 + px);
        oBase[o]             = (su0 + eps * sc0) * rden;
        oBase[o + plane]     = (su1 + eps * sc1) * rden;
        oBase[o + 2 * plane] = (su2 + eps * sc2) * rden;
    }
}

extern "C" void kernel_launch(void* const* d_in, const int* in_sizes, int n_in,
                              void* d_out, int out_size, void* d_ws, size_t ws_size,
                              hipStream_t stream) {
    (void)n_in; (void)out_size; (void)d_ws; (void)ws_size;
    const float* depth = (const float*)d_in[0];
    const float* color = (const float*)d_in[1];
    const float* mask  = (const float*)d_in[2];
    float* out = (float*)d_out;

    const int n_imgs = in_sizes[0] / (IMG_H * IMG_W);   // B*V = 8

    dim3 grid(IMG_W / TW, IMG_H / THT, n_imgs);
    dim3 block(NTHREADS, 1, 1);
    bilateral3x3_kernel<<<grid, block, 0, stream>>>(depth, color, mask, out);
}